// RNNDecoder_40106404610780
// MI455X (gfx1250) — compile-verified
//
#include <hip/hip_runtime.h>
#include <hip/hip_bf16.h>
#include <math.h>
#include <stdint.h>

// Problem dims (fixed by the reference)
#define B_  64
#define N_  32
#define T_  50
#define D_  4
#define K_  4
#define H_  256
#define E_  992          // N*(N-1)
#define F_  512          // 2*H
#define TS  49           // T-1

// LDS leading dims padded for bank-conflict-free column access (64 banks):
// 516 % 64 == 4, 260 % 64 == 4 -> rows 0..15 of a column land in distinct banks.
#define LDA 516
#define LDM 260

typedef __attribute__((ext_vector_type(2))) float v2f;
typedef __attribute__((ext_vector_type(8))) float v8f;

__device__ __forceinline__ v8f wmma_f32_k4(v2f a, v2f b, v8f c) {
  // 8 args: (neg_a, A, neg_b, B, c_mod, C, reuse_a, reuse_b)
  return __builtin_amdgcn_wmma_f32_16x16x4_f32(false, a, false, b, (short)0, c,
                                               false, false);
}

// CDNA5 async memory->LDS copy, 16 bytes per lane. VDST VGPR = LDS byte addr,
// VGPR offset + SGPR base (GVS mode). Tracked by ASYNCcnt.
__device__ __forceinline__ void async_copy16(unsigned lds_addr,
                                             unsigned long long saddr,
                                             unsigned byte_off) {
  asm volatile("global_load_async_to_lds_b128 %0, %1, %2"
               :: "v"(lds_addr), "v"(byte_off), "s"(saddr)
               : "memory");
}
__device__ __forceinline__ void async_wait0() {
  asm volatile("s_wait_asynccnt 0x0" ::: "memory");
}
__device__ __forceinline__ unsigned lds_off(const float* p) {
  return (unsigned)(uintptr_t)p;   // low 32 bits of flat shared ptr = LDS offset
}

// One 16x16 output tile: A tile lives in LDS (16 x kdim, leading dim lda),
// B matrix is global, packed K-pairs: Bp[((f>>1)*256 + col)*2 + (f&1)] = W[f][col].
// f32 16x16x4 fragment layout:
//   A: lane<16 -> row=lane, K = {f0, f0+1};  lane>=16 -> row=lane-16, K = {f0+2, f0+3}
//   B: symmetric (lane%16 = column, lane/16 selects K pair)
//   C: vgpr j -> (M = j + 8*(lane/16), N = lane%16)
__device__ __forceinline__ v8f gemm_tile(const float* sA, int lda,
                                         const float* __restrict__ Bp, int kdim,
                                         int n0, int lane) {
  const int row  = lane & 15;
  const int koff = (lane >> 4) << 1;          // 0 or 2 (even)
  const int col  = n0 + row;
  v8f acc = {};
  for (int f0 = 0; f0 < kdim; f0 += 4) {
    v2f a, b;
    a.x = sA[row * lda + f0 + koff];
    a.y = sA[row * lda + f0 + koff + 1];
    const int p = (f0 + koff) >> 1;           // K-pair index
    b = *(const v2f*)(Bp + ((p * 256 + col) << 1));   // single b64 load
    acc = wmma_f32_k4(a, b, acc);
  }
  return acc;
}

// ---- setup kernels ------------------------------------------------------

// one-hot (E,N) -> index array
__global__ void k_find_idx(const float* __restrict__ rel, int* __restrict__ idx) {
  int e = blockIdx.x * blockDim.x + threadIdx.x;
  if (e < E_) {
    int best = 0; float bv = rel[e * N_];
    for (int n = 1; n < N_; n++) { float v = rel[e * N_ + n]; if (v > bv) { bv = v; best = n; } }
    idx[e] = best;
  }
}

// Transpose + pack K-pairs: in (R=256 rows "h", C cols "f") row-major.
// out[((f>>1)*256 + h)*2 + (f&1)] = in[h*C + f]   (i.e. W[f][h], f-pairs adjacent)
__global__ void k_transpose_pack(const float* __restrict__ in, float* __restrict__ out,
                                 int R, int C) {
  int i = blockIdx.x * blockDim.x + threadIdx.x;
  if (i < R * C) {
    int r = i / C, c = i % C;
    out[(((c >> 1) * R + r) << 1) + (c & 1)] = in[i];
  }
}

// ---- per-step: edge message MLP + aggregation ---------------------------

__global__ __launch_bounds__(256) void k_msg(
    const float* __restrict__ hidden, const float* __restrict__ rel_type,
    const int* __restrict__ sidxg, const int* __restrict__ ridxg,
    const float* __restrict__ w1t, const float* __restrict__ b1,
    const float* __restrict__ w2t, const float* __restrict__ b2,
    float* __restrict__ agg) {
  __shared__ float sA[16 * LDA];      // pre_msg tile 16 x 512
  __shared__ float sM[16 * LDM];      // m1 tile 16 x 256
  __shared__ float sMsg[16 * LDM];    // weighted message accumulator
  __shared__ int   sidx[16], ridx[16];
  __shared__ float srt[16][3];

  const int tid = threadIdx.x;
  const int b   = blockIdx.x / (E_ / 16);
  const int e0  = (blockIdx.x % (E_ / 16)) * 16;

  if (tid < 16) { sidx[tid] = sidxg[e0 + tid]; ridx[tid] = ridxg[e0 + tid]; }
  if (tid < 48) {
    int m = tid / 3, kk = tid % 3;
    srt[m][kk] = rel_type[(b * E_ + e0 + m) * K_ + kk + 1] * (1.0f / 3.0f);
  }
  __syncthreads();

  // stage pre_msg = [hidden[send] , hidden[recv]] via async mem->LDS copies
  // 16 rows x 512 floats = 2048 chunks of 16B; 4-float chunks never straddle
  // the send/recv boundary (256 % 4 == 0) nor the padded row end.
  {
    const unsigned long long hb = (unsigned long long)(hidden + (size_t)b * N_ * H_);
    for (int c = tid; c < 16 * 128; c += 256) {
      int m  = c >> 7;
      int fq = (c & 127) << 2;                 // float col within pre_msg row
      int node = (fq < H_) ? sidx[m] : ridx[m];
      int col  = fq & (H_ - 1);
      async_copy16(lds_off(&sA[m * LDA + fq]), hb,
                   (unsigned)((node * H_ + col) * 4));
    }
  }
  for (int i = tid; i < 16 * H_; i += 256) {
    int m = i >> 8, h = i & 255;
    sMsg[m * LDM + h] = 0.0f;
  }
  async_wait0();
  __syncthreads();

  const int wave = tid >> 5, lane = tid & 31;
  const int row  = lane & 15;
  const int hofs = (lane >> 4) * 8;

  for (int kk = 0; kk < 3; kk++) {
    const float* W1  = w1t + kk * (F_ * H_);
    const float* W2  = w2t + kk * (H_ * H_);
    const float* bb1 = b1 + (kk + 1) * H_;
    const float* bb2 = b2 + (kk + 1) * H_;

    // GEMM1: (16 x 512) x (512 x 256) -> tanh -> sM
    for (int ct = 0; ct < 2; ct++) {
      int n0 = wave * 32 + ct * 16;
      v8f acc = gemm_tile(sA, LDA, W1, F_, n0, lane);
      float bias = bb1[n0 + row];
#pragma unroll
      for (int j = 0; j < 8; j++) {
        int m = j + hofs;
        sM[m * LDM + n0 + row] = tanhf(acc[j] + bias);
      }
    }
    __syncthreads();

    // GEMM2: (16 x 256) x (256 x 256) -> tanh -> weighted accumulate
    for (int ct = 0; ct < 2; ct++) {
      int n0 = wave * 32 + ct * 16;
      v8f acc = gemm_tile(sM, LDM, W2, H_, n0, lane);
      float bias = bb2[n0 + row];
#pragma unroll
      for (int j = 0; j < 8; j++) {
        int m = j + hofs;
        sMsg[m * LDM + n0 + row] += tanhf(acc[j] + bias) * srt[m][kk];
      }
    }
    __syncthreads();
  }

  // scatter-add to receiver nodes (includes /D)
  for (int i = tid; i < 16 * H_; i += 256) {
    int m = i >> 8, h = i & 255;
    atomicAdd(&agg[(b * N_ + ridx[m]) * H_ + h], sMsg[m * LDM + h] * 0.25f);
  }
}

// ---- per-step: GRU update + output MLP ----------------------------------

__global__ __launch_bounds__(256) void k_gru_out(
    const float* __restrict__ data, const float* __restrict__ agg,
    float* __restrict__ hidden,
    const float* __restrict__ hidrT, const float* __restrict__ hidiT,
    const float* __restrict__ hidhT,
    const float* __restrict__ in_r_w, const float* __restrict__ in_r_b,
    const float* __restrict__ in_i_w, const float* __restrict__ in_i_b,
    const float* __restrict__ in_n_w, const float* __restrict__ in_n_b,
    const float* __restrict__ out1T, const float* __restrict__ out1_b,
    const float* __restrict__ out2T, const float* __restrict__ out2_b,
    const float* __restrict__ out3_w, const float* __restrict__ out3_b,
    float* __restrict__ out, int t) {
  __shared__ float s0[16 * LDM];   // agg tile, then hnew, reused as GEMM A
  __shared__ float s1[16 * LDM];   // r-gate pre / mlp1
  __shared__ float s2[16 * LDM];   // i-gate pre / mlp2
  __shared__ float s3[16 * LDM];   // n-gate pre
  __shared__ float sIns[16][4];

  const int tid  = threadIdx.x;
  const int r0   = blockIdx.x * 16;          // row in flattened (B*N)
  const int wave = tid >> 5, lane = tid & 31;
  const int row  = lane & 15;
  const int hofs = (lane >> 4) * 8;

  if (tid < 64) {
    int m = tid >> 2, d = tid & 3;
    sIns[m][d] = data[((r0 + m) * T_ + t) * D_ + d];
  }
  // stage agg tile via async mem->LDS copies (16 x 256 floats = 1024 chunks)
  {
    const unsigned long long ab = (unsigned long long)(agg + (size_t)r0 * H_);
    for (int c = tid; c < 16 * 64; c += 256) {
      int m = c >> 6, h = (c & 63) << 2;
      async_copy16(lds_off(&s0[m * LDM + h]), ab, (unsigned)((m * H_ + h) * 4));
    }
  }
  async_wait0();
  __syncthreads();

  // three recurrent-gate GEMMs: agg @ hid_*_w^T
  for (int ct = 0; ct < 2; ct++) {
    int n0 = wave * 32 + ct * 16;
    v8f aR = gemm_tile(s0, LDM, hidrT, H_, n0, lane);
    v8f aI = gemm_tile(s0, LDM, hidiT, H_, n0, lane);
    v8f aN = gemm_tile(s0, LDM, hidhT, H_, n0, lane);
#pragma unroll
    for (int j = 0; j < 8; j++) {
      int m = j + hofs;
      s1[m * LDM + n0 + row] = aR[j];
      s2[m * LDM + n0 + row] = aI[j];
      s3[m * LDM + n0 + row] = aN[j];
    }
  }
  __syncthreads();

  // elementwise GRU update (D=4 input matmuls are tiny -> VALU)
  for (int i = tid; i < 16 * H_; i += 256) {
    int m = i >> 8, h = i & 255;
    float i0 = sIns[m][0], i1 = sIns[m][1], i2 = sIns[m][2], i3 = sIns[m][3];
    float pr = in_r_b[h] + i0 * in_r_w[h * 4 + 0] + i1 * in_r_w[h * 4 + 1]
                         + i2 * in_r_w[h * 4 + 2] + i3 * in_r_w[h * 4 + 3];
    float pi = in_i_b[h] + i0 * in_i_w[h * 4 + 0] + i1 * in_i_w[h * 4 + 1]
                         + i2 * in_i_w[h * 4 + 2] + i3 * in_i_w[h * 4 + 3];
    float pn = in_n_b[h] + i0 * in_n_w[h * 4 + 0] + i1 * in_n_w[h * 4 + 1]
                         + i2 * in_n_w[h * 4 + 2] + i3 * in_n_w[h * 4 + 3];
    float r  = 1.0f / (1.0f + expf(-(pr + s1[m * LDM + h])));
    float ii = 1.0f / (1.0f + expf(-(pi + s2[m * LDM + h])));
    float nn = tanhf(pn + r * s3[m * LDM + h]);
    float hold = hidden[(r0 + m) * H_ + h];
    float hnew = (1.0f - ii) * nn + ii * hold;
    hidden[(r0 + m) * H_ + h] = hnew;
    s0[m * LDM + h] = hnew;
  }
  __syncthreads();

  // output MLP layer 1: relu(hnew @ out1^T + b1)
  for (int ct = 0; ct < 2; ct++) {
    int n0 = wave * 32 + ct * 16;
    v8f acc = gemm_tile(s0, LDM, out1T, H_, n0, lane);
    float bias = out1_b[n0 + row];
#pragma unroll
    for (int j = 0; j < 8; j++) {
      int m = j + hofs;
      float v = acc[j] + bias;
      s1[m * LDM + n0 + row] = v > 0.0f ? v : 0.0f;
    }
  }
  __syncthreads();

  // output MLP layer 2: relu(pr @ out2^T + b2)
  for (int ct = 0; ct < 2; ct++) {
    int n0 = wave * 32 + ct * 16;
    v8f acc = gemm_tile(s1, LDM, out2T, H_, n0, lane);
    float bias = out2_b[n0 + row];
#pragma unroll
    for (int j = 0; j < 8; j++) {
      int m = j + hofs;
      float v = acc[j] + bias;
      s2[m * LDM + n0 + row] = v > 0.0f ? v : 0.0f;
    }
  }
  __syncthreads();

  // final projection to D=4 and skip connection
  if (tid < 64) {
    int m = tid >> 2, d = tid & 3;
    float acc = out3_b[d];
    for (int h = 0; h < H_; h++) acc += s2[m * LDM + h] * out3_w[d * H_ + h];
    out[((r0 + m) * TS + t) * D_ + d] = sIns[m][d] + acc;
  }
}

// ---- host ---------------------------------------------------------------

extern "C" void kernel_launch(void* const* d_in, const int* in_sizes, int n_in,
                              void* d_out, int out_size, void* d_ws, size_t ws_size,
                              hipStream_t stream) {
  (void)in_sizes; (void)n_in; (void)out_size; (void)ws_size;
  const float* data     = (const float*)d_in[0];
  const float* rel_type = (const float*)d_in[1];
  const float* rel_rec  = (const float*)d_in[2];
  const float* rel_send = (const float*)d_in[3];
  const float* msg_w1   = (const float*)d_in[4];
  const float* msg_b1   = (const float*)d_in[5];
  const float* msg_w2   = (const float*)d_in[6];
  const float* msg_b2   = (const float*)d_in[7];
  const float* hid_r_w  = (const float*)d_in[8];
  const float* hid_i_w  = (const float*)d_in[9];
  const float* hid_h_w  = (const float*)d_in[10];
  const float* in_r_w   = (const float*)d_in[11];
  const float* in_r_b   = (const float*)d_in[12];
  const float* in_i_w   = (const float*)d_in[13];
  const float* in_i_b   = (const float*)d_in[14];
  const float* in_n_w   = (const float*)d_in[15];
  const float* in_n_b   = (const float*)d_in[16];
  const float* out1_w   = (const float*)d_in[17];
  const float* out1_b   = (const float*)d_in[18];
  const float* out2_w   = (const float*)d_in[19];
  const float* out2_b   = (const float*)d_in[20];
  const float* out3_w   = (const float*)d_in[21];
  const float* out3_b   = (const float*)d_in[22];

  // workspace layout (~7.9 MB total)
  float* ws     = (float*)d_ws;
  float* hidden = ws;                       // B*N*H           = 524288
  float* agg    = hidden + 524288;          //                  524288
  float* w1t    = agg + 524288;             // 3*F*H          = 393216
  float* w2t    = w1t + 393216;             // 3*H*H          = 196608
  float* hidrT  = w2t + 196608;             // H*H            =  65536
  float* hidiT  = hidrT + 65536;
  float* hidhT  = hidiT + 65536;
  float* out1T  = hidhT + 65536;
  float* out2T  = out1T + 65536;
  int*   sidx   = (int*)(out2T + 65536);    // E ints
  int*   ridx   = sidx + E_;

  hipMemsetAsync(hidden, 0, 524288 * sizeof(float), stream);

  k_find_idx<<<(E_ + 255) / 256, 256, 0, stream>>>(rel_send, sidx);
  k_find_idx<<<(E_ + 255) / 256, 256, 0, stream>>>(rel_rec, ridx);

  for (int kk = 0; kk < 3; kk++) {   // SKIP_FIRST: use weight slices 1..3
    k_transpose_pack<<<(H_ * F_ + 255) / 256, 256, 0, stream>>>(
        msg_w1 + (size_t)(kk + 1) * H_ * F_, w1t + (size_t)kk * F_ * H_, H_, F_);
    k_transpose_pack<<<(H_ * H_ + 255) / 256, 256, 0, stream>>>(
        msg_w2 + (size_t)(kk + 1) * H_ * H_, w2t + (size_t)kk * H_ * H_, H_, H_);
  }
  k_transpose_pack<<<(H_ * H_ + 255) / 256, 256, 0, stream>>>(hid_r_w, hidrT, H_, H_);
  k_transpose_pack<<<(H_ * H_ + 255) / 256, 256, 0, stream>>>(hid_i_w, hidiT, H_, H_);
  k_transpose_pack<<<(H_ * H_ + 255) / 256, 256, 0, stream>>>(hid_h_w, hidhT, H_, H_);
  k_transpose_pack<<<(H_ * H_ + 255) / 256, 256, 0, stream>>>(out1_w, out1T, H_, H_);
  k_transpose_pack<<<(H_ * H_ + 255) / 256, 256, 0, stream>>>(out2_w, out2T, H_, H_);

  for (int t = 0; t < TS; t++) {
    hipMemsetAsync(agg, 0, 524288 * sizeof(float), stream);
    k_msg<<<B_ * (E_ / 16), 256, 0, stream>>>(hidden, rel_type, sidx, ridx,
                                              w1t, msg_b1, w2t, msg_b2, agg);
    k_gru_out<<<(B_ * N_) / 16, 256, 0, stream>>>(
        data, agg, hidden, hidrT, hidiT, hidhT,
        in_r_w, in_r_b, in_i_w, in_i_b, in_n_w, in_n_b,
        out1T, out1_b, out2T, out2_b, out3_w, out3_b, (float*)d_out, t);
  }
}